// KLDoBCELoss_25168508354919
// MI455X (gfx1250) — compile-verified
//
#include <hip/hip_runtime.h>
#include <math.h>

// Problem constants (match reference)
#define N_SAMP 1024
#define K_ROWS 17
#define W_X 384
#define W_Y 512
#define THR 3.0f
#define LBETA 10.0f

typedef float v4f __attribute__((ext_vector_type(4)));

// ---------- wave32 reductions (CDNA5 is wave32-only) ----------
__device__ __forceinline__ float wred_sum(float v) {
#pragma unroll
    for (int m = 16; m > 0; m >>= 1) v += __shfl_xor(v, m, 32);
    return v;
}
__device__ __forceinline__ float wred_max(float v) {
#pragma unroll
    for (int m = 16; m > 0; m >>= 1) v = fmaxf(v, __shfl_xor(v, m, 32));
    return v;
}
__device__ __forceinline__ int wred_and(int v) {
#pragma unroll
    for (int m = 16; m > 0; m >>= 1) v &= __shfl_xor(v, m, 32);
    return v;
}

// One block per sample n. 17 waves, one wave per row (n,k).
// pred stays register-resident (used in all 3 passes); gt is staged into LDS
// via CDNA5 async-to-LDS DMA (ASYNCcnt) and re-read per pass, halving VGPR
// pressure so more waves/SIMD cover the HBM latency of this streaming kernel.
template <int W>
__global__ __launch_bounds__(K_ROWS * 32)
void axis_kernel(const float* __restrict__ pred,
                 const float* __restrict__ gt,
                 const float* __restrict__ tw,
                 float* __restrict__ partial) {
    constexpr int C = W / 128;          // float4 chunks per lane
    const int n    = blockIdx.x;        // sample
    const int wave = threadIdx.x >> 5;  // k in [0, 17)
    const int lane = threadIdx.x & 31;

    __shared__ __align__(16) float s_gt[K_ROWS][W];
    __shared__ float s_loss[K_ROWS];
    __shared__ int   s_less[K_ROWS];

    if (threadIdx.x == 0) {
        // weight row is consumed at the end by wave 0 -> warm it early
        __builtin_prefetch(tw + (size_t)n * K_ROWS, 0, 0);
    }

    const size_t row = (size_t)n * K_ROWS + wave;
    const v4f* pr = reinterpret_cast<const v4f*>(pred + row * W);

    // ---- issue async gt-row copy: global -> LDS (128b per lane per chunk) ----
    {
        const unsigned long long gbase =
            (unsigned long long)(uintptr_t)(gt + row * W) + (unsigned)lane * 16u;
        const unsigned lbase =
            (unsigned)(uintptr_t)(&s_gt[wave][0]) + (unsigned)lane * 16u;
#pragma unroll
        for (int c = 0; c < C; ++c) {
            const unsigned long long ga = gbase + (unsigned)c * 512u;
            const unsigned la = lbase + (unsigned)c * 512u;
            asm volatile("global_load_async_to_lds_b128 %0, %1, off"
                         :: "v"(la), "v"(ga)
                         : "memory");
        }
    }

    // ---- pred loads overlap the async DMA ----
    v4f p[C];
#pragma unroll
    for (int c = 0; c < C; ++c)
        p[c] = __builtin_nontemporal_load(pr + c * 32 + lane);

    // pass 1a: pred-side stats while gt is still in flight
    float mp = -INFINITY;
    int ok = 1;
#pragma unroll
    for (int c = 0; c < C; ++c) {
#pragma unroll
        for (int j = 0; j < 4; ++j) {
            const float pv = p[c][j];
            mp = fmaxf(mp, pv);
            ok &= (pv < THR) ? 1 : 0;
        }
    }

    // ---- gt staged copy complete? (wave consumes only its own region) ----
    asm volatile("s_wait_asynccnt 0x0" ::: "memory");
    const v4f* gl = reinterpret_cast<const v4f*>(&s_gt[wave][0]);

    // pass 1b: max(10*gt) from LDS
    float mg = -INFINITY;
#pragma unroll
    for (int c = 0; c < C; ++c) {
        const v4f gv = gl[c * 32 + lane];
#pragma unroll
        for (int j = 0; j < 4; ++j) mg = fmaxf(mg, LBETA * gv[j]);
    }

    mp = wred_max(mp);
    mg = wred_max(mg);
    const int lesser = wred_and(ok);

    // pass 2: softmax denominators
    float sp = 0.f, sg = 0.f;
#pragma unroll
    for (int c = 0; c < C; ++c) {
        const v4f gv = gl[c * 32 + lane];
#pragma unroll
        for (int j = 0; j < 4; ++j) {
            sp += __expf(p[c][j] - mp);
            sg += __expf(fmaf(LBETA, gv[j], -mg));
        }
    }
    sp = wred_sum(sp);
    sg = wred_sum(sg);

    // per-row constants: cp = log-sum-exp(pred), off folds label normalization
    const float cp  = mp + __logf(sp);      // log_pt = p - cp
    const float off = -(mg + __logf(sg));   // log_lab = 10*g + off

    // pass 3: KLD and BCE accumulation (FMA-restructured)
    float kld = 0.f, bce = 0.f;
#pragma unroll
    for (int c = 0; c < C; ++c) {
        const v4f gv = gl[c * 32 + lane];
#pragma unroll
        for (int j = 0; j < 4; ++j) {
            const float pv = p[c][j];
            const float g  = gv[j];
            const float log_lab = fmaf(LBETA, g, off);     // log softmax(10*gt)
            const float lab     = __expf(log_lab);
            // xlogy(l,l) - l*log_pt = l * (log_lab - (p - cp))
            kld = fmaf(lab, (log_lab + cp) - pv, kld);
            // logaddexp(0, p) = max(p,0) + log(1 + exp(-|p|))
            const float e = __expf(-fabsf(pv));
            bce += fmaxf(pv, 0.f) + __logf(1.0f + e);
            bce = fmaf(-pv, g, bce);
        }
    }
    kld = wred_sum(kld);
    bce = wred_sum(bce);

    if (lane == 0) {
        s_loss[wave] = (lesser ? bce : kld) * (1.0f / (float)W);
        s_less[wave] = lesser;
    }
    __syncthreads();

    // wave 0: stable-argsort rank via mask+popcount, weighted sum
    if (wave == 0) {
        unsigned mask = 0;
#pragma unroll
        for (int i = 0; i < K_ROWS; ++i)
            mask |= (s_less[i] ? 1u : 0u) << i;
        const int numFalse = K_ROWS - __popc(mask);

        float contrib = 0.f;
        if (lane < K_ROWS) {
            const int trueBefore = __popc(mask & ((1u << lane) - 1u));
            const int pos = (mask >> lane & 1u) ? (numFalse + trueBefore)
                                                : (lane - trueBefore);
            contrib = s_loss[lane] * tw[(size_t)n * K_ROWS + pos];
        }
        contrib = wred_sum(contrib);
        if (lane == 0) partial[n] = contrib;
    }
}

// Deterministic tree reduction of the 2*N partials -> out[0] = sum / K
__global__ __launch_bounds__(256)
void finish_kernel(const float* __restrict__ px,
                   const float* __restrict__ py,
                   float* __restrict__ out) {
    __shared__ float red[256];
    float s = 0.f;
    for (int i = threadIdx.x; i < N_SAMP; i += 256) s += px[i] + py[i];
    red[threadIdx.x] = s;
    __syncthreads();
#pragma unroll
    for (int off = 128; off > 0; off >>= 1) {
        if (threadIdx.x < off) red[threadIdx.x] += red[threadIdx.x + off];
        __syncthreads();
    }
    if (threadIdx.x == 0) out[0] = red[0] * (1.0f / (float)K_ROWS);
}

extern "C" void kernel_launch(void* const* d_in, const int* in_sizes, int n_in,
                              void* d_out, int out_size, void* d_ws, size_t ws_size,
                              hipStream_t stream) {
    (void)in_sizes; (void)n_in; (void)out_size; (void)ws_size;
    const float* pred_x = (const float*)d_in[0];
    const float* pred_y = (const float*)d_in[1];
    const float* gt_x   = (const float*)d_in[2];
    const float* gt_y   = (const float*)d_in[3];
    const float* tw     = (const float*)d_in[4];

    float* px = (float*)d_ws;       // N_SAMP partials (x axis)
    float* py = px + N_SAMP;        // N_SAMP partials (y axis)

    axis_kernel<W_X><<<N_SAMP, K_ROWS * 32, 0, stream>>>(pred_x, gt_x, tw, px);
    axis_kernel<W_Y><<<N_SAMP, K_ROWS * 32, 0, stream>>>(pred_y, gt_y, tw, py);
    finish_kernel<<<1, 256, 0, stream>>>(px, py, (float*)d_out);
}